// Encoder_17772574671593
// MI455X (gfx1250) — compile-verified
//
#include <hip/hip_runtime.h>
#include <hip/hip_bf16.h>
#include <stddef.h>

// ---------------------------------------------------------------------------
// 2-layer LSTM encoder (B=256, T=1024, F=64, H1=16, H2=12) + BatchNorm
// Fused WMMA-f32 scan, one 16-batch-row tile per wave32, 16 workgroups.
// All weight B-fragments register-resident; zero-C first WMMA per chain.
// ---------------------------------------------------------------------------

typedef float v2f __attribute__((ext_vector_type(2)));
typedef float v8f __attribute__((ext_vector_type(8)));

#define T_STEPS 1024
#define BATCH   256
#define F_IN    64

__device__ __forceinline__ v8f wmma4(v2f a, v2f b, v8f c) {
  // D = A(16x4 f32) * B(4x16 f32) + C(16x16 f32)
  return __builtin_amdgcn_wmma_f32_16x16x4_f32(
      /*neg_a=*/false, a, /*neg_b=*/false, b,
      /*c_mod=*/(short)0, c, /*reuse_a=*/false, /*reuse_b=*/false);
}

__device__ __forceinline__ v8f zero8() {
  v8f r;
#pragma unroll
  for (int i = 0; i < 8; ++i) r[i] = 0.f;
  return r;
}

__device__ __forceinline__ float sigm(float v) {
  return __builtin_amdgcn_rcpf(1.f + __expf(-v));
}

__device__ __forceinline__ float selu(float v) {
  const float lam = 1.0507009873554805f;
  const float la  = 1.7580993408473766f;  // lambda * alpha
  return v > 0.f ? lam * v : la * (__expf(v) - 1.f);
}

// LDS strides (floats): transposed weights [n][k], k-stride padded.
#define S_W1 68   // W1t: 64 n-rows x K=64 (staging only; dead after preload)
#define S_SM 20   // U1t/W2t/U2t: 64 n-rows x K=16 ; h buffers 16 x 16

__global__ void __launch_bounds__(32)
lstm2_scan_kernel(const float* __restrict__ x,
                  const float* __restrict__ w1, const float* __restrict__ u1,
                  const float* __restrict__ b1,
                  const float* __restrict__ w2, const float* __restrict__ u2,
                  const float* __restrict__ b2,
                  float* __restrict__ out)
{
  __shared__ float W1t[64 * S_W1];
  __shared__ float U1t[64 * S_SM];
  __shared__ float W2t[64 * S_SM];
  __shared__ float U2t[64 * S_SM];
  __shared__ float H1B[16 * S_SM];
  __shared__ float H2B[16 * S_SM];

  const int lane = threadIdx.x;        // 0..31
  const int half = lane >> 4;          // 0: K {0,1} / rows 0..7 ; 1: K {2,3} / rows 8..15
  const int col  = lane & 15;          // N index for B/C/D, M index for A-frags
  const int b0   = blockIdx.x * 16;    // batch tile base

  // ---- stage weights into LDS (transposed: [n][k]); pad layer-2 with zeros
  for (int i = lane; i < 64 * S_SM; i += 32) { W2t[i] = 0.f; U2t[i] = 0.f; }
  for (int i = lane; i < 64 * 64; i += 32) { int k = i >> 6, n = i & 63; W1t[n * S_W1 + k] = w1[i]; }
  for (int i = lane; i < 16 * 64; i += 32) { int k = i >> 6, n = i & 63; U1t[n * S_SM + k] = u1[i]; }
  for (int i = lane; i < 16 * 48; i += 32) {
    int k = i / 48, c = i - 48 * k, g = c / 12, j = c - 12 * g;
    W2t[(16 * g + j) * S_SM + k] = w2[i];
  }
  for (int i = lane; i < 12 * 48; i += 32) {
    int k = i / 48, c = i - 48 * k, g = c / 12, j = c - 12 * g;
    U2t[(16 * g + j) * S_SM + k] = u2[i];
  }
  __syncthreads();

  // ---- ALL loop-invariant B-fragments -> registers (B[k][n], k = 4*kc+2*half)
  v2f W1f[4][16];
#pragma unroll
  for (int g = 0; g < 4; ++g)
#pragma unroll
    for (int kc = 0; kc < 16; ++kc)
      W1f[g][kc] = *(const v2f*)&W1t[(16 * g + col) * S_W1 + 4 * kc + 2 * half];

  v2f U1f[4][4], W2f[4][4], U2f[4][4];
#pragma unroll
  for (int g = 0; g < 4; ++g)
#pragma unroll
    for (int kc = 0; kc < 4; ++kc) {
      const int n = 16 * g + col, k = 4 * kc + 2 * half;
      U1f[g][kc] = *(const v2f*)&U1t[n * S_SM + k];
      W2f[g][kc] = *(const v2f*)&W2t[n * S_SM + k];
      U2f[g][kc] = *(const v2f*)&U2t[n * S_SM + k];
    }

  float b1g[4], b2g[4];
#pragma unroll
  for (int g = 0; g < 4; ++g) {
    b1g[g] = b1[16 * g + col];
    b2g[g] = (col < 12) ? b2[12 * g + col] : 0.f;
  }

  v2f h1f[4] = {}, h2f[4] = {};           // recurrent A-fragments (start = 0)
  v8f c1 = zero8(), c2 = zero8();
  v8f h2v = zero8();

  // x A-fragment base: row m = col, element offset 4*kc + 2*half
  const float* xrow = x + ((size_t)(b0 + col) * T_STEPS) * F_IN + 2 * half;

  for (int t = 0; t < T_STEPS; ++t) {
    const float* xt = xrow + (size_t)t * F_IN;
    if (t + 1 < T_STEPS) __builtin_prefetch(xt + F_IN, 0, 0);  // next step's row

    v2f xf[16];
#pragma unroll
    for (int kc = 0; kc < 16; ++kc) xf[kc] = *(const v2f*)&xt[4 * kc];

    // ---- layer 1: z = x_t*W1 + h1*U1 (+b1 in epilogue); gate tiles i,f,g,o
    v8f zi = wmma4(xf[0], W1f[0][0], zero8());
    v8f zf = wmma4(xf[0], W1f[1][0], zero8());
    v8f zg = wmma4(xf[0], W1f[2][0], zero8());
    v8f zo = wmma4(xf[0], W1f[3][0], zero8());
#pragma unroll
    for (int kc = 1; kc < 16; ++kc) {
      v2f a = xf[kc];
      zi = wmma4(a, W1f[0][kc], zi);
      zf = wmma4(a, W1f[1][kc], zf);
      zg = wmma4(a, W1f[2][kc], zg);
      zo = wmma4(a, W1f[3][kc], zo);
    }
#pragma unroll
    for (int kc = 0; kc < 4; ++kc) {
      v2f a = h1f[kc];
      zi = wmma4(a, U1f[0][kc], zi);
      zf = wmma4(a, U1f[1][kc], zf);
      zg = wmma4(a, U1f[2][kc], zg);
      zo = wmma4(a, U1f[3][kc], zo);
    }
    v8f h1v;
#pragma unroll
    for (int e = 0; e < 8; ++e) {
      float ig = sigm(zi[e] + b1g[0]);
      float fg = sigm(zf[e] + b1g[1]);
      float og = sigm(zo[e] + b1g[3]);
      float cc = fg * c1[e] + ig * selu(zg[e] + b1g[2]);
      c1[e] = cc;
      h1v[e] = og * selu(cc);
    }
    // C-layout -> LDS -> A-layout fragments (row = e + 8*half, col = lane&15)
#pragma unroll
    for (int e = 0; e < 8; ++e) H1B[(e + 8 * half) * S_SM + col] = h1v[e];
    __syncthreads();
#pragma unroll
    for (int kc = 0; kc < 4; ++kc)
      h1f[kc] = *(const v2f*)&H1B[col * S_SM + 4 * kc + 2 * half];

    // ---- layer 2: y = h1*W2p + h2*U2p (+b2p); 48 cols padded into 4x16 tiles
    v8f yi = wmma4(h1f[0], W2f[0][0], zero8());
    v8f yf = wmma4(h1f[0], W2f[1][0], zero8());
    v8f yg = wmma4(h1f[0], W2f[2][0], zero8());
    v8f yo = wmma4(h1f[0], W2f[3][0], zero8());
#pragma unroll
    for (int kc = 1; kc < 4; ++kc) {
      v2f a = h1f[kc];
      yi = wmma4(a, W2f[0][kc], yi);
      yf = wmma4(a, W2f[1][kc], yf);
      yg = wmma4(a, W2f[2][kc], yg);
      yo = wmma4(a, W2f[3][kc], yo);
    }
#pragma unroll
    for (int kc = 0; kc < 4; ++kc) {
      v2f a = h2f[kc];
      yi = wmma4(a, U2f[0][kc], yi);
      yf = wmma4(a, U2f[1][kc], yf);
      yg = wmma4(a, U2f[2][kc], yg);
      yo = wmma4(a, U2f[3][kc], yo);
    }
#pragma unroll
    for (int e = 0; e < 8; ++e) {
      float ig = sigm(yi[e] + b2g[0]);
      float fg = sigm(yf[e] + b2g[1]);
      float og = sigm(yo[e] + b2g[3]);
      float cc = fg * c2[e] + ig * selu(yg[e] + b2g[2]);
      c2[e] = cc;
      h2v[e] = og * selu(cc);               // pad cols (>=12) stay exactly 0
      H2B[(e + 8 * half) * S_SM + col] = h2v[e];
    }
    __syncthreads();
#pragma unroll
    for (int kc = 0; kc < 4; ++kc)
      h2f[kc] = *(const v2f*)&H2B[col * S_SM + 4 * kc + 2 * half];
  }

  // ---- final h2 (pre-BatchNorm) -> out[B,12]
  if (col < 12) {
#pragma unroll
    for (int e = 0; e < 8; ++e)
      out[(size_t)(b0 + e + 8 * half) * 12 + col] = h2v[e];
  }
}

// ---------------------------------------------------------------------------
// BatchNorm over batch axis (training-mode stats), in place on out[256,12].
// ---------------------------------------------------------------------------
__global__ void __launch_bounds__(256)
bn_kernel(float* __restrict__ out,
          const float* __restrict__ gamma, const float* __restrict__ beta)
{
  __shared__ float s_mu[12], s_rs[12], s_bt[12];
  const int t = threadIdx.x;
  if (t < 12) {
    float s = 0.f, s2 = 0.f;
    for (int b = 0; b < BATCH; ++b) {
      float v = out[b * 12 + t];
      s += v; s2 += v * v;
    }
    const float mu  = s * (1.f / BATCH);
    const float var = s2 * (1.f / BATCH) - mu * mu;
    s_mu[t] = mu;
    s_rs[t] = rsqrtf(var + 1e-3f) * gamma[t];
    s_bt[t] = beta[t];
  }
  __syncthreads();
  const int b = t;  // one row per thread
#pragma unroll
  for (int j = 0; j < 12; ++j) {
    float v = out[b * 12 + j];
    out[b * 12 + j] = (v - s_mu[j]) * s_rs[j] + s_bt[j];
  }
}

extern "C" void kernel_launch(void* const* d_in, const int* in_sizes, int n_in,
                              void* d_out, int out_size, void* d_ws, size_t ws_size,
                              hipStream_t stream) {
  (void)in_sizes; (void)n_in; (void)out_size; (void)d_ws; (void)ws_size;
  const float* x     = (const float*)d_in[0];
  const float* w1    = (const float*)d_in[1];
  const float* u1    = (const float*)d_in[2];
  const float* b1    = (const float*)d_in[3];
  const float* w2    = (const float*)d_in[4];
  const float* u2    = (const float*)d_in[5];
  const float* b2    = (const float*)d_in[6];
  const float* gamma = (const float*)d_in[7];
  const float* beta  = (const float*)d_in[8];
  float* out = (float*)d_out;

  lstm2_scan_kernel<<<BATCH / 16, 32, 0, stream>>>(x, w1, u1, b1, w2, u2, b2, out);
  bn_kernel<<<1, BATCH, 0, stream>>>(out, gamma, beta);
}